// RNN_61125974557431
// MI455X (gfx1250) — compile-verified
//
#include <hip/hip_runtime.h>
#include <hip/hip_bf16.h>

typedef __attribute__((ext_vector_type(2))) float v2f;
typedef __attribute__((ext_vector_type(8))) float v8f;

namespace {
constexpr int kB = 2048;
constexpr int kT = 512;
constexpr int kF = 64;   // input features = K of x@Wx
constexpr int kH = 16;   // hidden
constexpr int kG = 64;   // 4*H gate width
}

__device__ __forceinline__ float sigmoidf_fast(float v) {
    return 1.0f / (1.0f + __expf(-v));
}

// One wave (32 lanes) owns a 16-row batch tile for the whole T=512 scan.
// z[16x64] = x_t[16x64] @ Wx[64x64] + h[16x16] @ Wh[16x64] + b, all via
// V_WMMA_F32_16X16X4_F32 (full f32 precision through the recurrence).
// The x@Wx partial is software-pipelined one step ahead so only the 4-deep
// h@Wh WMMA chain + gate math + transpose sit on the serial critical path,
// and the matrix pipe stays busy (WMMA co-executes with gate VALU).
__global__ __launch_bounds__(32) void lstm_fused_wmma(
    const float* __restrict__ x,
    const float* __restrict__ Wx,
    const float* __restrict__ Wh,
    const float* __restrict__ b,
    const float* __restrict__ W2,
    const float* __restrict__ b2,
    const float* __restrict__ W3,
    const float* __restrict__ b3,
    const float* __restrict__ Wo,
    const float* __restrict__ bo,
    float* __restrict__ out)
{
    // Wx pre-packed in WMMA-B fragment order: [kc(16)][n(4)][lane(32)][2 floats]
    __shared__ float lds_wx[16 * 4 * 32 * 2];  // 16 KB
    __shared__ float lds_h[16 * 16];           // h D-layout -> A-layout transpose

    const int lane = threadIdx.x;   // 0..31
    const int lrow = lane & 15;     // row / col within 16-wide tile
    const int lhi  = lane >> 4;     // lane-half selects K pair (ISA 7.12.2)
    const int row0 = blockIdx.x * 16;

    // ---- pre-pack Wx B-fragments into LDS (one-time, 128 strided loads) ----
    // B layout (4x16 f32, 2 VGPRs): vgpr v, lane l: K = kc*4 + 2*(l>>4) + v,
    // N = l&15.  Pack the two K values contiguously -> hot loop ds_load_b64.
    #pragma unroll
    for (int kc = 0; kc < 16; ++kc) {
        #pragma unroll
        for (int n = 0; n < 4; ++n) {
            const int kbase = kc * 4 + 2 * lhi;
            v2f frag;
            frag.x = Wx[(kbase + 0) * kG + n * 16 + lrow];
            frag.y = Wx[(kbase + 1) * kG + n * 16 + lrow];
            *(v2f*)&lds_wx[((kc * 4 + n) * 32 + lane) * 2] = frag;
        }
    }

    // ---- Wh B-fragments stay register-resident (32 VGPRs) ----
    v2f WhB[4][4];
    #pragma unroll
    for (int kc = 0; kc < 4; ++kc) {
        #pragma unroll
        for (int n = 0; n < 4; ++n) {
            const int kbase = kc * 4 + 2 * lhi;
            WhB[kc][n].x = Wh[(kbase + 0) * kG + n * 16 + lrow];
            WhB[kc][n].y = Wh[(kbase + 1) * kG + n * 16 + lrow];
        }
    }

    // bias folded into the accumulator seed: column-only dependence (N = lrow),
    // so all 8 C/D slots of a gate tile share one value -> splatted v8f.
    v8f binit[4];
    #pragma unroll
    for (int n = 0; n < 4; ++n) {
        const float bvn = b[n * 16 + lrow];
        #pragma unroll
        for (int e = 0; e < 8; ++e) binit[n][e] = bvn;
    }

    // LSTM state in C/D layout: vgpr e, lanes 0-15: (M=e, N=lane);
    // lanes 16-31: (M=e+8, N=lane-16). N indexes the hidden dim here.
    v8f c8 = {0.f, 0.f, 0.f, 0.f, 0.f, 0.f, 0.f, 0.f};
    v8f h8 = {0.f, 0.f, 0.f, 0.f, 0.f, 0.f, 0.f, 0.f};
    v2f hA[4];   // h in A layout for next step's h@Wh
    #pragma unroll
    for (int kc = 0; kc < 4; ++kc) { hA[kc].x = 0.f; hA[kc].y = 0.f; }

    __syncthreads();

    // A-layout x row base for this lane: lane holds row M = lrow of the tile
    const float* xrow = x + ((size_t)(row0 + lrow)) * kT * kF + 2 * lhi;

    // zx(t) = x_t @ Wx + b in C/D layout (independent of the recurrence)
    auto compute_zx = [&](int t, v8f zx[4]) {
        const float* xp = xrow + (size_t)t * kF;
        v2f xA[16];
        #pragma unroll
        for (int kc = 0; kc < 16; ++kc)
            xA[kc] = *(const v2f*)(xp + kc * 4);
        #pragma unroll
        for (int n = 0; n < 4; ++n) {
            v8f acc = binit[n];
            #pragma unroll
            for (int kc = 0; kc < 16; ++kc) {
                const v2f wxf = *(const v2f*)&lds_wx[((kc * 4 + n) * 32 + lane) * 2];
                acc = __builtin_amdgcn_wmma_f32_16x16x4_f32(
                    false, xA[kc], false, wxf, (short)0, acc, false, false);
            }
            zx[n] = acc;
        }
    };

    v8f zx[4];
    compute_zx(0, zx);

    for (int t = 0; t < kT; ++t) {
        // ---- critical path: z = zx + h @ Wh (only 4 WMMAs deep) ----
        v8f z[4];
        #pragma unroll
        for (int n = 0; n < 4; ++n) {
            v8f acc = zx[n];
            #pragma unroll
            for (int kc = 0; kc < 4; ++kc) {
                acc = __builtin_amdgcn_wmma_f32_16x16x4_f32(
                    false, hA[kc], false, WhB[kc][n], (short)0, acc, false, false);
            }
            z[n] = acc;
        }

        // ---- off critical path: issue next step's x@Wx now; these 64 WMMAs
        // co-execute with the gate VALU/TRANS work below ----
        if (t + 1 < kT) {
            __builtin_prefetch(xrow + (size_t)(t + 2) * kF, 0, 3);
            compute_zx(t + 1, zx);
        }

        // gate math (Keras order i,f,g,o; activation=relu on g and on c in h)
        #pragma unroll
        for (int e = 0; e < 8; ++e) {
            const float ig = sigmoidf_fast(z[0][e]);
            const float fg = sigmoidf_fast(z[1][e]);
            const float gg = fmaxf(z[2][e], 0.0f);
            const float og = sigmoidf_fast(z[3][e]);
            const float cn = fg * c8[e] + ig * gg;
            c8[e] = cn;
            h8[e] = og * fmaxf(cn, 0.0f);
        }

        // transpose h: C/D layout -> A layout via LDS (single-wave WG:
        // barriers are near-free, ds ops stay ordered per wave)
        __syncthreads();
        #pragma unroll
        for (int e = 0; e < 8; ++e)
            lds_h[(e + 8 * lhi) * 16 + lrow] = h8[e];
        __syncthreads();
        #pragma unroll
        for (int kc = 0; kc < 4; ++kc)
            hA[kc] = *(const v2f*)&lds_h[lrow * 16 + kc * 4 + 2 * lhi];
    }

    // ---- tiny head MLP on h_T: 16 -> 8 -> 4 -> 1, one batch row per lane ----
    __syncthreads();   // lds_h holds h_T[m][n] from the last iteration's store
    if (lane < 16) {
        const int r = lane;
        float hv[16];
        #pragma unroll
        for (int k = 0; k < 16; ++k) hv[k] = lds_h[r * 16 + k];

        float x2[8];
        #pragma unroll
        for (int j = 0; j < 8; ++j) {
            float a = b2[j];
            #pragma unroll
            for (int k = 0; k < 16; ++k) a += hv[k] * W2[k * 8 + j];
            x2[j] = fmaxf(a, 0.0f);
        }
        float x3[4];
        #pragma unroll
        for (int m = 0; m < 4; ++m) {
            float a = b3[m];
            #pragma unroll
            for (int j = 0; j < 8; ++j) a += x2[j] * W3[j * 4 + m];
            x3[m] = fmaxf(a, 0.0f);
        }
        float o = bo[0];
        #pragma unroll
        for (int m = 0; m < 4; ++m) o += x3[m] * Wo[m];
        out[row0 + r] = sigmoidf_fast(o);
    }
}

extern "C" void kernel_launch(void* const* d_in, const int* in_sizes, int n_in,
                              void* d_out, int out_size, void* d_ws, size_t ws_size,
                              hipStream_t stream) {
    const float* x  = (const float*)d_in[0];
    const float* Wx = (const float*)d_in[1];
    const float* Wh = (const float*)d_in[2];
    const float* b  = (const float*)d_in[3];
    const float* W2 = (const float*)d_in[4];
    const float* b2 = (const float*)d_in[5];
    const float* W3 = (const float*)d_in[6];
    const float* b3 = (const float*)d_in[7];
    const float* Wo = (const float*)d_in[8];
    const float* bo = (const float*)d_in[9];
    float* out = (float*)d_out;

    (void)in_sizes; (void)n_in; (void)out_size; (void)d_ws; (void)ws_size;

    dim3 grid(kB / 16);   // 128 single-wave workgroups, 16 batch rows each
    dim3 block(32);
    lstm_fused_wmma<<<grid, block, 0, stream>>>(x, Wx, Wh, b, W2, b2, W3, b3,
                                                Wo, bo, out);
}